// CoordinateDescent_29454885716088
// MI455X (gfx1250) — compile-verified
//
#include <hip/hip_runtime.h>

typedef __attribute__((ext_vector_type(2))) float v2f;
typedef __attribute__((ext_vector_type(8))) float v8f;
typedef int v4i_vs __attribute__((vector_size(16)));   // matches builtin param type

// Problem constants (fixed by the reference's setup_inputs)
#define BATCH 16
#define MDIM  2048
#define NDIM  2048
#define RDIM  64

#define TILE_K 64
#define LDS_PAD 4

// ---- CDNA5 async global->LDS path (guarded: falls back to VGPR staging) ----
#if defined(__HIP_DEVICE_COMPILE__) && __has_builtin(__builtin_amdgcn_global_load_async_to_lds_b128) && __has_builtin(__builtin_amdgcn_s_wait_asynccnt)
#define USE_ASYNC 1
#else
#define USE_ASYNC 0
#endif

#if USE_ASYNC
typedef __attribute__((address_space(3))) v4i_vs* lds_v4i_ptr;
__device__ __forceinline__ void async_copy16(const float* gsrc, float* ldst) {
    // generic LDS address: low 32 bits are the LDS byte offset (ISA aperture rule)
    lds_v4i_ptr dst = (lds_v4i_ptr)(unsigned int)(unsigned long long)(char*)ldst;
    __builtin_amdgcn_global_load_async_to_lds_b128((v4i_vs*)gsrc, dst, 0, 0);
}
__device__ __forceinline__ void async_wait0() { __builtin_amdgcn_s_wait_asynccnt(0); }
#else
__device__ __forceinline__ void async_wait0() {}
#endif

// ---------------------------------------------------------------------------
// Tiled fp32 WMMA GEMM:  Out[b, i, 0:64] = sum_k Xop[b, i, k] * W[b, k, 0:64]
//   TRANS_A == false:  Xop[i,k] = X[i*ldx + k]   (A-tile staged i-major)
//   TRANS_A == true :  Xop[i,k] = X[k*ldx + i]   (A-tile staged k-major so the
//                                                 16B async destination is
//                                                 contiguous; frag reads
//                                                 transpose out of LDS)
// W is [K x 64] row-major, staged k-transposed (wsT[r][k]) for b64 frag loads.
// 128 threads = 4 waves; each wave computes a (16*RW) x 64 strip with RW*4
// accumulators of v_wmma_f32_16x16x4_f32 (B-frags reused across RW).
// I must be a multiple of 64*RW, K a multiple of 64 (true at all call sites).
// ---------------------------------------------------------------------------
template <bool TRANS_A, int RW>
__global__ void __launch_bounds__(128)
gemm_wmma_f32(const float* __restrict__ X, const float* __restrict__ W,
              float* __restrict__ Out, int K, int ldx,
              long xbs, long wbs, long obs)
{
    constexpr int ITILE = 64 * RW;
    constexpr int XROWS = TRANS_A ? TILE_K : ITILE;
    constexpr int XCOLS = (TRANS_A ? ITILE : TILE_K) + LDS_PAD;

    __shared__ float xsbuf[XROWS][XCOLS];
    __shared__ float wsT[RDIM][TILE_K + LDS_PAD];   // wsT[r][k] = W[k][r]

    const int batch = blockIdx.y;
    const int i0    = blockIdx.x * ITILE;
    const float* Xb = X + (long)batch * xbs;
    const float* Wb = W + (long)batch * wbs;
    float*       Ob = Out + (long)batch * obs;

    const int tid  = threadIdx.x;
    const int wave = tid >> 5;
    const int lane = tid & 31;

    v8f c[RW][4] = {};

    for (int k0 = 0; k0 < K; k0 += TILE_K) {
        // ---- stage A tile (async direct-to-LDS when available) ----
        if constexpr (!TRANS_A) {
            // xsbuf[ii][kk]; global contiguous along k
            for (int e = tid; e < ITILE * (TILE_K / 4); e += 128) {
                const int ii = e >> 4;             // / (TILE_K/4)
                const int k4 = (e & 15) << 2;
                const float* src = Xb + (long)(i0 + ii) * ldx + (k0 + k4);
#if USE_ASYNC
                async_copy16(src, &xsbuf[ii][k4]);
#else
                float4 val = *(const float4*)src;
                xsbuf[ii][k4 + 0] = val.x; xsbuf[ii][k4 + 1] = val.y;
                xsbuf[ii][k4 + 2] = val.z; xsbuf[ii][k4 + 3] = val.w;
#endif
            }
        } else {
            // xsbuf[kk][ii]; global contiguous along i
            constexpr int I4 = ITILE / 4;
            for (int e = tid; e < TILE_K * I4; e += 128) {
                const int kk = e / I4;
                const int i4 = (e % I4) << 2;
                const float* src = Xb + (long)(k0 + kk) * ldx + (i0 + i4);
#if USE_ASYNC
                async_copy16(src, &xsbuf[kk][i4]);
#else
                float4 val = *(const float4*)src;
                xsbuf[kk][i4 + 0] = val.x; xsbuf[kk][i4 + 1] = val.y;
                xsbuf[kk][i4 + 2] = val.z; xsbuf[kk][i4 + 3] = val.w;
#endif
            }
        }
        // ---- stage W tile transposed: wsT[r][kk] = W[k0+kk][r] (small) ----
        for (int e = tid; e < TILE_K * (RDIM / 4); e += 128) {
            const int kk = e >> 4;
            const int r4 = (e & 15) << 2;
            float4 val = *(const float4*)(Wb + (long)(k0 + kk) * RDIM + r4);
            wsT[r4 + 0][kk] = val.x; wsT[r4 + 1][kk] = val.y;
            wsT[r4 + 2][kk] = val.z; wsT[r4 + 3][kk] = val.w;
        }
        async_wait0();
        __syncthreads();

        // ISA A-layout (16x4 f32): lanes 0-15 hold K=0,1 ; lanes 16-31 K=2,3
        const int lane15 = lane & 15;
        const int khalf  = (lane >> 4) << 1;       // 0 or 2
        const int arow0  = wave * (16 * RW) + lane15;
#pragma unroll
        for (int kk = 0; kk < TILE_K; kk += 4) {
            v2f af[RW];
#pragma unroll
            for (int rw = 0; rw < RW; ++rw) {
                const int ar = arow0 + 16 * rw;
                if constexpr (!TRANS_A) {
                    af[rw].x = xsbuf[ar][kk + khalf];
                    af[rw].y = xsbuf[ar][kk + khalf + 1];
                } else {
                    af[rw].x = xsbuf[kk + khalf][ar];
                    af[rw].y = xsbuf[kk + khalf + 1][ar];
                }
            }
            v2f bf[4];
#pragma unroll
            for (int t = 0; t < 4; ++t) {
                bf[t].x = wsT[t * 16 + lane15][kk + khalf];
                bf[t].y = wsT[t * 16 + lane15][kk + khalf + 1];
            }
#pragma unroll
            for (int rw = 0; rw < RW; ++rw)
#pragma unroll
                for (int t = 0; t < 4; ++t)
                    c[rw][t] = __builtin_amdgcn_wmma_f32_16x16x4_f32(
                        false, af[rw], false, bf[t], (short)0, c[rw][t], false, false);
        }
        __syncthreads();
    }

    // C/D layout: lane&15 = N column; M row = jj + 8*(lane>>4)
    const int col = lane & 15;
#pragma unroll
    for (int rw = 0; rw < RW; ++rw) {
        const int mbase = i0 + wave * (16 * RW) + rw * 16 + ((lane >> 4) << 3);
#pragma unroll
        for (int jj = 0; jj < 8; ++jj) {
            float* orow = Ob + (long)(mbase + jj) * RDIM;
            orow[ 0 + col] = c[rw][0][jj];
            orow[16 + col] = c[rw][1][jj];
            orow[32 + col] = c[rw][2][jj];
            orow[48 + col] = c[rw][3][jj];
        }
    }
}

// ---------------------------------------------------------------------------
// Coordinate descent: one thread per (batch,row). u-row lives in 64 VGPRs;
// Gram matrix b (symmetric, 64x64) lives in LDS. Sequential over r.
//   u[r] = soft_thr(a[r] - (u . b[:,r] - u[r]*b[r][r]), l1) / (b[r][r]+l2+eps)
// ---------------------------------------------------------------------------
__global__ void __launch_bounds__(256)
cd_rows(const float* __restrict__ a, const float* __restrict__ bmat,
        const float* __restrict__ u_in, float* __restrict__ u_out,
        int rows, float l1, float l2)
{
    __shared__ float bs[RDIM * RDIM];
    const int batch = blockIdx.y;
    const float* bb = bmat + (long)batch * RDIM * RDIM;
    for (int i = threadIdx.x; i < RDIM * RDIM; i += 256) bs[i] = bb[i];
    __syncthreads();

    const int row = blockIdx.x * 256 + threadIdx.x;
    const long base = ((long)batch * rows + row) * RDIM;
    const float* arow = a + base;
    const float* urow = u_in + base;
    float*       orow = u_out + base;

    float ur[RDIM];
    float ar[RDIM];
#pragma unroll
    for (int j = 0; j < RDIM; j += 4) {
        float4 t = *(const float4*)(urow + j);
        ur[j] = t.x; ur[j + 1] = t.y; ur[j + 2] = t.z; ur[j + 3] = t.w;
        float4 s = *(const float4*)(arow + j);
        ar[j] = s.x; ar[j + 1] = s.y; ar[j + 2] = s.z; ar[j + 3] = s.w;
    }

    for (int r = 0; r < RDIM; ++r) {
        const float* brow = &bs[r * RDIM];    // symmetric: b[:,r] == b[r,:]
        const float brr = brow[r];
        float t = 0.0f;
#pragma unroll
        for (int j = 0; j < RDIM; ++j) t = fmaf(ur[j], brow[j], t);
        t -= ur[r] * brr;
        float num = ar[r] - t;
        float st  = fmaxf(fabsf(num) - l1, 0.0f);
        st = copysignf(st, num);
        ur[r] = st / (brr + l2 + 1e-16f);
    }

#pragma unroll
    for (int j = 0; j < RDIM; j += 4) {
        float4 t = make_float4(ur[j], ur[j + 1], ur[j + 2], ur[j + 3]);
        *(float4*)(orow + j) = t;
    }
}

// ---------------------------------------------------------------------------
extern "C" void kernel_launch(void* const* d_in, const int* in_sizes, int n_in,
                              void* d_out, int out_size, void* d_ws, size_t ws_size,
                              hipStream_t stream)
{
    (void)in_sizes; (void)n_in; (void)out_size; (void)ws_size;
    const float* x = (const float*)d_in[0];   // [B, M, N]
    const float* u = (const float*)d_in[1];   // [B, M, R]
    const float* v = (const float*)d_in[2];   // [B, N, R]

    float* u_out = (float*)d_out;                                 // [B, M, R]
    float* v_out = u_out + (size_t)BATCH * MDIM * RDIM;           // [B, N, R]

    float* a    = (float*)d_ws;                                   // [B, max(M,N), R]
    float* bmat = a + (size_t)BATCH * MDIM * RDIM;                // [B, R, R]

    // regularizers (reference: L2_U=L2_V=0.01, L1_RATIO=0.1)
    const float l1_u = 0.01f * 0.1f * (float)NDIM;
    const float l2_u = 0.01f * 0.9f * (float)NDIM;
    const float l1_v = 0.01f * 0.1f * (float)MDIM;
    const float l2_v = 0.01f * 0.9f * (float)MDIM;

    // ---- factor 0: u ----
    // a = x @ v                      [B, M, R]
    gemm_wmma_f32<false, 2><<<dim3(MDIM / 128, BATCH), 128, 0, stream>>>(
        x, v, a, NDIM, NDIM,
        (long)MDIM * NDIM, (long)NDIM * RDIM, (long)MDIM * RDIM);
    // bmat = v^T @ v                 [B, R, R]
    gemm_wmma_f32<true, 1><<<dim3(1, BATCH), 128, 0, stream>>>(
        v, v, bmat, NDIM, RDIM,
        (long)NDIM * RDIM, (long)NDIM * RDIM, (long)RDIM * RDIM);
    // u_new
    cd_rows<<<dim3(MDIM / 256, BATCH), 256, 0, stream>>>(
        a, bmat, u, u_out, MDIM, l1_u, l2_u);

    // ---- factor 1: v (uses new u) ----
    // a = x^T @ u_new                [B, N, R]
    gemm_wmma_f32<true, 2><<<dim3(NDIM / 128, BATCH), 128, 0, stream>>>(
        x, u_out, a, MDIM, NDIM,
        (long)MDIM * NDIM, (long)MDIM * RDIM, (long)NDIM * RDIM);
    // bmat = u_new^T @ u_new         [B, R, R]
    gemm_wmma_f32<true, 1><<<dim3(1, BATCH), 128, 0, stream>>>(
        u_out, u_out, bmat, MDIM, RDIM,
        (long)MDIM * RDIM, (long)MDIM * RDIM, (long)RDIM * RDIM);
    // v_new
    cd_rows<<<dim3(NDIM / 256, BATCH), 256, 0, stream>>>(
        a, bmat, v, v_out, NDIM, l1_v, l2_v);
}